// ThreeBlindMice_59777354825883
// MI455X (gfx1250) — compile-verified
//
#include <hip/hip_runtime.h>
#include <hip/hip_bf16.h>
#include <math.h>

// ---------------------------------------------------------------------------
// ThreeBlindMice on MI455X (gfx1250, wave32, WMMA, TDM)
//
// Cost model:
//  - GEMV g = w_ih[18432x13500] @ cw : 995 MB fp32 streamed once -> ~43us at
//    23.3 TB/s. WMMA f32 16x16x4, one wave per 16-row tile, b128 loads feeding
//    two WMMAs each; x vector staged into LDS by the Tensor Data Mover.
//  - 3-layer T=512 hidden-9 LSTM: strictly serial -> software-pipelined across
//    3 waves (layer l lags layer l-1 by one step): 514 steps instead of 1536,
//    36 gates spread over 32 lanes (18-MAC chains, register-resident weights).
//  - Everything else is tiny (preprocess, gates, bitonic top-k + logsoftmax).
// ---------------------------------------------------------------------------

typedef float v2f __attribute__((ext_vector_type(2)));
typedef float v4f __attribute__((ext_vector_type(4)));
typedef float v8f __attribute__((ext_vector_type(8)));
typedef unsigned int u32x4 __attribute__((ext_vector_type(4)));
typedef int i32x4 __attribute__((ext_vector_type(4)));
typedef int i32x8 __attribute__((ext_vector_type(8)));

#define S_SYM   500
#define N_SMP   3
#define F_RAW   20
#define NF      11
#define KB      9
#define H_IN    13500          // S*N*9
#define H_OUT   4608           // 512*9
#define G_ROWS  18432          // 4*H_OUT
#define OUT_K   1500

#if __has_builtin(__builtin_amdgcn_tensor_load_to_lds) && \
    __has_builtin(__builtin_amdgcn_s_wait_tensorcnt)
#define USE_TDM 1
#else
#define USE_TDM 0
#endif

__device__ __forceinline__ float sigm(float x) { return 1.0f / (1.0f + expf(-x)); }

// ---------------------------------------------------------------------------
// Kernel 1: instance-norm -> rfft(20) -> bilinear(9x11x11) -> wax 3x3 matmul
//           -> tanh -> cw[13500].  One thread per (s, n): 1500 tasks.
// ---------------------------------------------------------------------------
__global__ void __launch_bounds__(128)
preprocess_kernel(const float* __restrict__ candles,   // [N,S,F]
                  const float* __restrict__ wax,       // [S,N,9]
                  const float* __restrict__ blw,       // [9,11,11]
                  const float* __restrict__ blb,       // [9]
                  float* __restrict__ cw) {            // [13500]
    int tid = blockIdx.x * blockDim.x + threadIdx.x;
    if (tid >= S_SYM * N_SMP) return;
    int s = tid / N_SMP, n = tid % N_SMP;

    float xv[F_RAW];
    const float* src = candles + ((size_t)n * S_SYM + s) * F_RAW;
    float mu = 0.f;
#pragma unroll
    for (int f = 0; f < F_RAW; ++f) { xv[f] = src[f]; mu += xv[f]; }
    mu *= (1.0f / F_RAW);
    float var = 0.f;
#pragma unroll
    for (int f = 0; f < F_RAW; ++f) { float d = xv[f] - mu; var += d * d; }
    var *= (1.0f / F_RAW);
    float inv = rsqrtf(var + 1e-9f);
#pragma unroll
    for (int f = 0; f < F_RAW; ++f) xv[f] = (xv[f] - mu) * inv;

    // rfft, 20 -> 11 complex bins: X[k] = sum_t x[t] * e^{-2*pi*i*k*t/20}
    float re[NF], im[NF];
#pragma unroll
    for (int k = 0; k < NF; ++k) {
        float sr = 0.f, si = 0.f;
#pragma unroll
        for (int t = 0; t < F_RAW; ++t) {
            float ang = -6.28318530717958647692f * (float)(k * t) * (1.0f / F_RAW);
            float sn, cs;
            __sincosf(ang, &sn, &cs);
            sr += xv[t] * cs;
            si += xv[t] * sn;
        }
        re[k] = sr; im[k] = si;
    }

    float bil[KB];
#pragma unroll
    for (int kk = 0; kk < KB; ++kk) {
        const float* wk = blw + kk * NF * NF;
        float acc = blb[kk];
#pragma unroll
        for (int j = 0; j < NF; ++j) {
            float t = 0.f;
#pragma unroll
            for (int i = 0; i < NF; ++i) t += re[i] * wk[i * NF + j];
            acc += t * im[j];
        }
        bil[kk] = acc;
    }

    const float* wx = wax + (size_t)tid * 9;
    float* outp = cw + (size_t)tid * 9;
#pragma unroll
    for (int i = 0; i < 3; ++i)
#pragma unroll
        for (int k = 0; k < 3; ++k) {
            float a = 0.f;
#pragma unroll
            for (int j = 0; j < 3; ++j) a += wx[i * 3 + j] * bil[j * 3 + k];
            outp[i * 3 + k] = tanhf(a);
        }
}

// ---------------------------------------------------------------------------
// Kernel 2: g[18432] = W[18432 x 13500] @ x via v_wmma_f32_16x16x4_f32.
//
// ISA 16x4 f32 A layout: lane m%16 = row m; VGPR j at lane-half h = K slot
// 2h+j. K order inside a WMMA is a free permutation (pure accumulation), so
// each lane b128-loads W[row, k0+4h .. +3] and the 4 floats feed TWO WMMAs:
//   wmma1 slots (h,j) <- k0+4h+j      (elements 0,1)
//   wmma2 slots (h,j) <- k0+4h+2+j    (elements 2,3)
// B is x broadcast over all 16 N columns (same slot mapping), read from LDS.
// x[13500] is DMA'd into LDS once per workgroup by the Tensor Data Mover.
// D column 0 lives in lanes 0 (M=0..7) and 16 (M=8..15).
// ---------------------------------------------------------------------------
__global__ void __launch_bounds__(128)
gemv_wmma_kernel(const float* __restrict__ W,
                 const float* __restrict__ x,
                 float* __restrict__ g) {
    __shared__ float xs[H_IN];                       // 54 KB of 320 KB WGP LDS

#if USE_TDM
    if (threadIdx.x < 32) {                          // wave 0 issues the DMA
        unsigned long long ga = (unsigned long long)(const void*)x;
        unsigned int lds_off = (unsigned int)(unsigned long long)(void*)xs;
        u32x4 g0;
        g0[0] = 1u;                                  // count=1, user desc
        g0[1] = lds_off;                             // lds_addr (bytes)
        g0[2] = (unsigned int)(ga & 0xFFFFFFFFull);  // global_addr[31:0]
        g0[3] = (unsigned int)((ga >> 32) & 0x1FFFFFFull) | (2u << 30); // +type=2
        i32x8 g1;
        g1[0] = (int)(2u << 16);                     // data_size=2 -> 4 bytes
        g1[1] = (int)((unsigned)H_IN << 16);         // tensor_dim0[15:0]
        g1[2] = (int)(1u << 16);                     // dim0 hi=0, tensor_dim1=1
        g1[3] = (int)((unsigned)H_IN << 16);         // dim1 hi=0, tile_dim0=13500
        g1[4] = 1;                                   // tile_dim1=1, tile_dim2=0
        g1[5] = H_IN;                                // tensor_dim0_stride lo
        g1[6] = 0; g1[7] = 0;
        i32x4 gz4 = {0, 0, 0, 0};                    // groups 2/3 unused (<=2D)
        i32x8 gz8 = {0, 0, 0, 0, 0, 0, 0, 0};
        __builtin_amdgcn_tensor_load_to_lds(g0, g1, gz4, gz4, gz8, 0);
        __builtin_amdgcn_s_wait_tensorcnt(0);
    }
#else
    for (int i = threadIdx.x; i < H_IN; i += 128) xs[i] = x[i];
#endif
    __syncthreads();

    const int lane = threadIdx.x & 31;
    const int wave = blockIdx.x * 4 + (threadIdx.x >> 5);
    const int m0   = wave * 16;
    const int half = lane >> 4;
    const int r    = lane & 15;

    const float* wbase = W + (size_t)(m0 + r) * H_IN + 4 * half;
    const float* xbase = xs + 4 * half;

    v8f acc = {0.f, 0.f, 0.f, 0.f, 0.f, 0.f, 0.f, 0.f};

#pragma unroll 4
    for (int k0 = 0; k0 < H_IN - 4; k0 += 8) {       // 1687 iterations
        __builtin_prefetch(wbase + k0 + 1024, 0, 0); // global_prefetch_b8
        v4f wv = *(const v4f*)(wbase + k0);          // global_load_b128
        v4f xv = *(const v4f*)(xbase + k0);          // ds_load_b128
        v2f a0 = {wv[0], wv[1]}, b0 = {xv[0], xv[1]};
        v2f a1 = {wv[2], wv[3]}, b1 = {xv[2], xv[3]};
        acc = __builtin_amdgcn_wmma_f32_16x16x4_f32(false, a0, false, b0,
                                                    (short)0, acc, false, false);
        acc = __builtin_amdgcn_wmma_f32_16x16x4_f32(false, a1, false, b1,
                                                    (short)0, acc, false, false);
    }
    {   // K tail: 13496..13499 with the plain b64 mapping
        const int k0 = H_IN - 4;
        v2f a = *(const v2f*)(W + (size_t)(m0 + r) * H_IN + k0 + 2 * half);
        v2f b = *(const v2f*)(xs + k0 + 2 * half);
        acc = __builtin_amdgcn_wmma_f32_16x16x4_f32(false, a, false, b,
                                                    (short)0, acc, false, false);
    }

    if (r == 0) {                                    // lanes 0 and 16: N == 0
        float* go = g + m0 + half * 8;               // M = VGPR + 8*half
#pragma unroll
        for (int j = 0; j < 8; ++j) go[j] = acc[j];
    }
}

// ---------------------------------------------------------------------------
// Kernel 3: LSTMCell gates with zero initial state:
//   c = sigmoid(i) * tanh(g);  h = sigmoid(o) * tanh(c)   (f gate unused)
// ---------------------------------------------------------------------------
__global__ void __launch_bounds__(256)
gates_kernel(const float* __restrict__ g,
             const float* __restrict__ bih,
             const float* __restrict__ bhh,
             float* __restrict__ h) {
    int i = blockIdx.x * blockDim.x + threadIdx.x;
    if (i >= H_OUT) return;
    float gi = g[i]             + bih[i]             + bhh[i];
    float gg = g[2 * H_OUT + i] + bih[2 * H_OUT + i] + bhh[2 * H_OUT + i];
    float go = g[3 * H_OUT + i] + bih[3 * H_OUT + i] + bhh[3 * H_OUT + i];
    float c  = sigm(gi) * tanhf(gg);
    h[i] = sigm(go) * tanhf(c);
}

// ---------------------------------------------------------------------------
// Kernel 4: 3-layer output LSTM, T=512, hidden=9, software-pipelined.
// Wave l owns layer l (3 waves). Layer l at step s processes t = s - l, so
// the whole chain completes in 514 barrier-separated steps instead of 1536.
// Within a wave, the 36 gate rows are spread over 32 lanes (rows 32..35 are a
// second accumulator on lanes 0..3), weights register-resident, h broadcast
// with __shfl. Sequence buffers live in LDS (4 x 18 KB).
// ---------------------------------------------------------------------------
__global__ void __launch_bounds__(96)
lstm3_kernel(const float* __restrict__ hin,     // [4608]
             const float* __restrict__ lwih,    // [3,36,9]
             const float* __restrict__ lwhh,    // [3,36,9]
             const float* __restrict__ lbih,    // [3,36]
             const float* __restrict__ lbhh,    // [3,36]
             float* __restrict__ seq_out) {     // [4608]
    __shared__ float buf[4 * H_OUT];            // buf[0]=input, buf[l+1]=layer l out

    const int tid  = threadIdx.x;
    const int wave = tid >> 5;                  // layer index 0..2
    const int lane = tid & 31;

    for (int i = tid; i < H_OUT; i += 96) buf[i] = hin[i];

    // Register-resident weights: row1 = lane (rows 0..31), row2 = 32+lane (lanes 0..3)
    const float* wiL = lwih + wave * 36 * 9;
    const float* whL = lwhh + wave * 36 * 9;
    float wi1[9], wh1[9], wi2[9], wh2[9];
#pragma unroll
    for (int p = 0; p < 9; ++p) { wi1[p] = wiL[lane * 9 + p]; wh1[p] = whL[lane * 9 + p]; }
    float b1 = lbih[wave * 36 + lane] + lbhh[wave * 36 + lane];
    float b2 = 0.f;
    if (lane < 4) {
#pragma unroll
        for (int p = 0; p < 9; ++p) { wi2[p] = wiL[(32 + lane) * 9 + p]; wh2[p] = whL[(32 + lane) * 9 + p]; }
        b2 = lbih[wave * 36 + 32 + lane] + lbhh[wave * 36 + 32 + lane];
    } else {
#pragma unroll
        for (int p = 0; p < 9; ++p) { wi2[p] = 0.f; wh2[p] = 0.f; }
    }

    const float* inb = buf + wave * H_OUT;
    float* outb      = buf + (wave + 1) * H_OUT;

    float hr = 0.f, cr = 0.f;
    __syncthreads();

    for (int s = 0; s < 512 + 2; ++s) {
        int t = s - wave;                        // wave-uniform pipeline stage
        if (t >= 0 && t < 512) {
            float hp[9];
#pragma unroll
            for (int p = 0; p < 9; ++p) hp[p] = __shfl(hr, p, 32);
            float xr[9];
            const float* xin = inb + t * 9;
#pragma unroll
            for (int p = 0; p < 9; ++p) xr[p] = xin[p];

            float a1 = b1, a2 = b2;              // rows lane and 32+lane
#pragma unroll
            for (int p = 0; p < 9; ++p) a1 += wi1[p] * xr[p] + wh1[p] * hp[p];
#pragma unroll
            for (int p = 0; p < 9; ++p) a2 += wi2[p] * xr[p] + wh2[p] * hp[p];

            // gather gates for hidden unit j == lane (torch order i,f,g,o)
            float gi  = a1;                               // row j       (<9)
            float gf  = __shfl(a1, 9 + lane, 32);         // row 9+j
            float gg  = __shfl(a1, 18 + lane, 32);        // row 18+j
            float goA = __shfl(a1, 27 + lane, 32);        // row 27+j (j<5)
            float goB = __shfl(a2, lane - 5, 32);         // row 32+(j-5) (j>=5)
            float go  = (lane < 5) ? goA : goB;

            if (lane < 9) {
                cr = sigm(gf) * cr + sigm(gi) * tanhf(gg);
                hr = sigm(go) * tanhf(cr);
                outb[t * 9 + lane] = hr;
            }
        }
        __syncthreads();
    }

    for (int i = tid; i < H_OUT; i += 96) seq_out[i] = buf[3 * H_OUT + i];
}

// ---------------------------------------------------------------------------
// Kernel 5: top-1500-of-4608 (values only, descending) + per-row log_softmax.
// Single-workgroup bitonic sort of 8192 (padded with -inf) in 32 KB LDS.
// ---------------------------------------------------------------------------
__global__ void __launch_bounds__(1024)
topk_logsoftmax_kernel(const float* __restrict__ flat,  // [4608]
                       float* __restrict__ outp) {      // [1500]
    __shared__ float sv[8192];
    __shared__ float rowLse[3];
    const int tid = threadIdx.x;

    for (int i = tid; i < 8192; i += 1024)
        sv[i] = (i < H_OUT) ? flat[i] : -3.0e38f;
    __syncthreads();

    for (int k = 2; k <= 8192; k <<= 1) {
        for (int j = k >> 1; j > 0; j >>= 1) {
            for (int i = tid; i < 8192; i += 1024) {
                int ixj = i ^ j;
                if (ixj > i) {
                    bool up = ((i & k) == 0);
                    float a = sv[i], b = sv[ixj];
                    if ((a > b) == up) { sv[i] = b; sv[ixj] = a; }
                }
            }
            __syncthreads();
        }
    }

    if (tid < 3) {                              // desc[p] = sv[8191 - p]
        int base = 8191 - tid * 500;
        float m = sv[base];                     // row max = first desc element
        float ssum = 0.f;
        for (int i = 0; i < 500; ++i) ssum += expf(sv[base - i] - m);
        rowLse[tid] = m + logf(ssum);
    }
    __syncthreads();

    for (int q = tid; q < OUT_K; q += 1024) {
        int r = q / 500, i = q % 500;
        outp[q] = sv[8191 - (r * 500 + i)] - rowLse[r];
    }
}

// ---------------------------------------------------------------------------
extern "C" void kernel_launch(void* const* d_in, const int* in_sizes, int n_in,
                              void* d_out, int out_size, void* d_ws, size_t ws_size,
                              hipStream_t stream) {
    const float* candles = (const float*)d_in[0];   // [3,500,20]
    const float* wax     = (const float*)d_in[1];   // [500,3,9]
    const float* bl_w    = (const float*)d_in[2];   // [9,11,11]
    const float* bl_b    = (const float*)d_in[3];   // [9]
    const float* w_ih    = (const float*)d_in[4];   // [18432,13500]
    // d_in[5] = w_hh: unused (h0 == 0)
    const float* b_ih    = (const float*)d_in[6];   // [18432]
    const float* b_hh    = (const float*)d_in[7];   // [18432]
    const float* lw_ih   = (const float*)d_in[8];   // [3,36,9]
    const float* lw_hh   = (const float*)d_in[9];   // [3,36,9]
    const float* lb_ih   = (const float*)d_in[10];  // [3,36]
    const float* lb_hh   = (const float*)d_in[11];  // [3,36]
    float* outp = (float*)d_out;                    // [1500]

    char* ws = (char*)d_ws;
    float* cw  = (float*)(ws);                          // 13500 floats
    float* g   = (float*)(ws + 54000);                  // 18432 floats
    float* h   = (float*)(ws + 54000 + 73728);          // 4608 floats
    float* seq = (float*)(ws + 54000 + 73728 + 18432);  // 4608 floats

    preprocess_kernel<<<12, 128, 0, stream>>>(candles, wax, bl_w, bl_b, cw);
    gemv_wmma_kernel<<<G_ROWS / 16 / 4, 128, 0, stream>>>(w_ih, cw, g);
    gates_kernel<<<H_OUT / 256, 256, 0, stream>>>(g, b_ih, b_hh, h);
    lstm3_kernel<<<1, 96, 0, stream>>>(h, lw_ih, lw_hh, lb_ih, lb_hh, seq);
    topk_logsoftmax_kernel<<<1, 1024, 0, stream>>>(seq, outp);
}